// AtomFeature_43954695308036
// MI455X (gfx1250) — compile-verified
//
#include <hip/hip_runtime.h>
#include <math.h>

// Only check builtin availability in the DEVICE pass; the host pass (x86)
// never has amdgcn builtins and must not trip this.
#if defined(__HIP_DEVICE_COMPILE__) && !__has_builtin(__builtin_amdgcn_wmma_f32_16x16x4_f32)
#error "wmma_f32_16x16x4_f32 builtin not available on this toolchain (device pass)"
#endif

typedef __attribute__((ext_vector_type(2))) float v2f;
typedef __attribute__((ext_vector_type(8))) float v8f;

#define NB 2
#define NL 256
#define NA 12
#define NTOT 3072            // NL*NA
#define KNB 30               // neighbours
#define NC 32                // embedding dim
#define BIGD 1000000.0f
#define SENT 0x7F7FFFFFu     // FLT_MAX bits: sentinel key-hi for masked pairs

// ---------------------------------------------------------------------------
// Kernel 1: atom_coords copy, atom_mask, graph-norm encode.
// mean/var depend only on (b,c); derived in closed form from emb_table + nres.
// ---------------------------------------------------------------------------
__global__ __launch_bounds__(256)
void encode_kernel(const float* __restrict__ coords, const int* __restrict__ mask,
                   const float* __restrict__ table, const float* __restrict__ scale,
                   const float* __restrict__ shift, float* __restrict__ out) {
    int b   = blockIdx.x;
    int tid = threadIdx.x;

    __shared__ int   s_nres;
    __shared__ float s_mean[NC], s_coef[NC];

    if (tid == 0) s_nres = 0;
    __syncthreads();
    atomicAdd(&s_nres, mask[b * NL + tid]);   // one residue per thread (NL==256)
    __syncthreads();

    int   nres = s_nres;
    float fcnt = fmaxf((float)(NA * nres), 1.0f);

    if (tid < NC) {
        int c = tid;
        float S = 0.f;
        #pragma unroll
        for (int a = 0; a < NA; ++a) S += table[a * NC + c];
        float mean = (float)nres * S / fcnt;
        float q = 0.f;
        #pragma unroll
        for (int a = 0; a < NA; ++a) { float d = table[a * NC + c] - mean; q += d * d; }
        float varnum = (float)nres * q + (float)(NTOT - NA * nres) * mean * mean;
        float stdv   = sqrtf(varnum / fcnt + 1e-5f);
        s_mean[c] = mean;
        s_coef[c] = scale[c] / stdv;
    }
    __syncthreads();

    float* out_coords = out;                          // NB*NTOT*3
    float* out_mask   = out + NB * NTOT * 3;          // NB*NTOT
    float* out_enc    = out_mask + NB * NTOT;         // NB*NTOT*NC

    for (int i = tid; i < NTOT * 3; i += 256)
        out_coords[b * NTOT * 3 + i] = coords[b * NTOT * 3 + i];

    for (int n = tid; n < NTOT; n += 256) {
        int m = mask[b * NL + n / NA];
        out_mask[b * NTOT + n] = (float)m;
        int a = n % NA;
        long base = ((long)b * NTOT + n) * NC;
        #pragma unroll
        for (int c = 0; c < NC; ++c) {
            float v = m ? ((table[a * NC + c] - s_mean[c]) * s_coef[c] + shift[c]) : 0.f;
            out_enc[base + c] = v;
        }
    }
}

// ---------------------------------------------------------------------------
// Kernel 2: pairwise distances via V_WMMA_F32_16X16X4_F32 + fused LDS top-30.
// One wave (32 threads) per 16-row tile.  d^2 = r_i^2 + r_j^2 - 2 x_i.x_j:
//   A row  = (-2x,-2y,-2z, 1)   (16x4)
//   B col  = ( x,  y,  z, r_j^2) (4x16)
//   C      =  r_i^2 broadcast per row  -> D = squared distance tile.
// Selection is done on d^2 (sqrt is monotone); sqrt applied to the 30 winners.
// ---------------------------------------------------------------------------
__global__ __launch_bounds__(32)
void dist_topk_kernel(const float* __restrict__ coords, const int* __restrict__ mask,
                      float* __restrict__ out_dist, float* __restrict__ out_idx) {
    int blk  = blockIdx.x;            // 0 .. NB*(NTOT/16)-1
    int b    = blk / (NTOT / 16);
    int i0   = (blk % (NTOT / 16)) * 16;
    const float* ac = coords + (long)b * NTOT * 3;
    const int*   mk = mask   + b * NL;

    int lane = threadIdx.x;
    int half = lane >> 4;             // 0: rows 0-7 / K0,K1   1: rows 8-15 / K2,K3
    int lm   = lane & 15;

    // key = (float_bits(d^2 or SENT) << 32) | col  -> ascending == (dist, idx)
    __shared__ unsigned long long lists[16][16][KNB];   // [row][colLane][slot] 60 KB
    __shared__ float s_mres[NL];                        // residue mask as float, 1 KB
    __shared__ int   s_heads[16][16];                   // merge heads, 1 KB

    // residue masks -> LDS (8 per lane)
    #pragma unroll
    for (int k = 0; k < NL / 32; ++k) {
        int res = lane * (NL / 32) + k;
        s_mres[res] = (float)mk[res];
    }
    #pragma unroll
    for (int r = 0; r < 8; ++r) {
        int row = r + 8 * half;
        for (int t = 0; t < KNB; ++t) lists[row][lm][t] = ~0ull;
    }

    // ---- A operand (16x4 f32 layout: VGPR0 = K0|K2, VGPR1 = K1|K3) ----
    int rowA = i0 + lm;
    v2f Aop;
    if (half == 0) { Aop.x = -2.f * ac[rowA * 3 + 0]; Aop.y = -2.f * ac[rowA * 3 + 1]; }
    else           { Aop.x = -2.f * ac[rowA * 3 + 2]; Aop.y = 1.0f; }

    // ---- C operand: r_i^2 per row; also cache row-valid flags ----
    v8f  Cop;
    bool vrow[8];
    #pragma unroll
    for (int r = 0; r < 8; ++r) {
        int row = i0 + r + 8 * half;
        float x = ac[row * 3], y = ac[row * 3 + 1], z = ac[row * 3 + 2];
        Cop[r]  = x * x + y * y + z * z;
        vrow[r] = mk[row / NA] != 0;
    }

    unsigned long long curmax[8];
    #pragma unroll
    for (int r = 0; r < 8; ++r) curmax[r] = ~0ull;

    __syncthreads();

    // incremental col/12: col = jt*16 + lm
    int q  = (lm >= NA) ? 1 : 0;
    int rr = (lm >= NA) ? lm - NA : lm;

    for (int jt = 0; jt < NTOT / 16; ++jt) {
        int col = jt * 16 + lm;
        // ---- B operand (4x16 f32: VGPR0 = K0|K1 rows, VGPR1 = K2|K3 rows) ----
        v2f Bop;
        float cx = ac[col * 3], cy = ac[col * 3 + 1], cz = ac[col * 3 + 2];
        if (half == 0) { Bop.x = cx; Bop.y = cz; }
        else           { Bop.x = cy; Bop.y = cx * cx + cy * cy + cz * cz; }

        v8f D = __builtin_amdgcn_wmma_f32_16x16x4_f32(
                    /*neg_a=*/false, Aop, /*neg_b=*/false, Bop,
                    /*c_mod=*/(short)0, Cop, /*reuse_a=*/false, /*reuse_b=*/false);

        bool vj = s_mres[q] != 0.f;
        // advance col/12 for next tile (col += 16)
        { int t2 = rr + 4; bool ov = t2 >= NA; q += ov ? 2 : 1; rr = ov ? t2 - NA : t2; }

        #pragma unroll
        for (int r = 0; r < 8; ++r) {
            int row = r + 8 * half;
            unsigned int hib = (vrow[r] && vj) ? __float_as_uint(fmaxf(D[r], 0.f))
                                               : SENT;
            unsigned long long key =
                ((unsigned long long)hib << 32) | (unsigned int)col;
            if (key < curmax[r]) {
                unsigned long long* lst = &lists[row][lm][0];
                int p = KNB - 1;
                while (p > 0 && lst[p - 1] > key) { lst[p] = lst[p - 1]; --p; }
                lst[p] = key;
                curmax[r] = lst[KNB - 1];
            }
        }
    }
    __syncthreads();

    // ---- merge 16 sorted lists per row; lane m handles row m ----
    if (lane < 16) {
        int row  = lane;
        int grow = i0 + row;
        bool rowvalid = mk[grow / NA] != 0;
        #pragma unroll
        for (int c = 0; c < 16; ++c) s_heads[row][c] = 0;
        long base_o = ((long)b * NTOT + grow) * KNB;
        for (int t = 0; t < KNB; ++t) {
            unsigned long long best = ~0ull;
            int bc = 0;
            for (int c = 0; c < 16; ++c) {
                int h = s_heads[row][c];
                if (h < KNB) {
                    unsigned long long k = lists[row][c][h];
                    if (k < best) { best = k; bc = c; }
                }
            }
            s_heads[row][bc]++;
            unsigned int hib = (unsigned int)(best >> 32);
            int idx = (int)(best & 0xFFFFFFFFull);
            float dist = (hib == SENT) ? BIGD
                                       : sqrtf(__uint_as_float(hib) + 1e-6f);
            if (!rowvalid) { dist = BIGD; idx = 0; }   // pad rule
            out_dist[base_o + t] = dist;
            out_idx [base_o + t] = (float)idx;
        }
    }
}

// ---------------------------------------------------------------------------
extern "C" void kernel_launch(void* const* d_in, const int* in_sizes, int n_in,
                              void* d_out, int out_size, void* d_ws, size_t ws_size,
                              hipStream_t stream) {
    const float* coords = (const float*)d_in[0];
    const int*   mask   = (const int*)  d_in[1];
    const float* table  = (const float*)d_in[2];
    const float* scale  = (const float*)d_in[3];
    const float* shift  = (const float*)d_in[4];
    float* out = (float*)d_out;

    // output layout (flat, return order):
    // coords: NB*NTOT*3 | mask: NB*NTOT | encode: NB*NTOT*NC | dist: NB*NTOT*K | idx: NB*NTOT*K
    float* out_dist = out + NB * NTOT * 3 + NB * NTOT + NB * NTOT * NC;
    float* out_idx  = out_dist + NB * NTOT * KNB;

    encode_kernel<<<NB, 256, 0, stream>>>(coords, mask, table, scale, shift, out);
    dist_topk_kernel<<<NB * (NTOT / 16), 32, 0, stream>>>(coords, mask, out_dist, out_idx);
}